// RNN_300647711426
// MI455X (gfx1250) — compile-verified
//
#include <hip/hip_runtime.h>
#include <hip/hip_bf16.h>
#include <math.h>

// Problem constants (from reference): L=4096, B=32, D_IN=128, D_LAT=512
#define LSEQ   4096
#define BATCH  32
#define DIN    128
#define DLAT   512
#define MTOT   (LSEQ * BATCH)      // 131072 rows of flattened X
#define CH     (BATCH * DLAT)      // 16384 channels (b, j)
#define CHUNK  128                 // scan chunk length
#define NCHUNK (LSEQ / CHUNK)      // 32 chunks

typedef __attribute__((ext_vector_type(2))) float v2f;
typedef __attribute__((ext_vector_type(8))) float v8f;

// ---------------------------------------------------------------------------
// Kernel 1: U = X @ wx + bh   (fp32 WMMA 16x16x4, U written into d_out)
// Block = 256 threads (8 waves). Block tile: 32 rows (M) x 64 cols (N).
// Wave w: mSub = w & 1 (2 row-subtiles), nSub = w >> 1 (4 col-subtiles).
// ---------------------------------------------------------------------------
#define A_STRIDE 132   // 128 + 4 pad: conflict-free ds_load_b64 A-fragment reads

__global__ __launch_bounds__(256)
void rnn_gemm_wmma(const float* __restrict__ X,
                   const float* __restrict__ wx,
                   const float* __restrict__ bh,
                   float* __restrict__ U) {
    __shared__ __align__(16) float As[32 * A_STRIDE];

    const int tid   = threadIdx.x;
    const int mBase = blockIdx.x * 32;        // global row base of block tile
    const int nBase = blockIdx.y * 64;        // global col base of block tile

    // Stage A tile (32 rows x 128 K) into LDS, coalesced float4 loads.
    {
        const float4* Xv = (const float4*)(X + (size_t)mBase * DIN);
        #pragma unroll
        for (int q = tid; q < 32 * (DIN / 4); q += 256) {
            const int row = q >> 5;           // q / 32
            const int c4  = q & 31;           // q % 32
            float4 v = Xv[q];
            *(float4*)&As[row * A_STRIDE + c4 * 4] = v;
        }
    }
    __syncthreads();

    const int wave = tid >> 5;
    const int lane = tid & 31;
    const int half = lane >> 4;               // 0: lanes 0-15, 1: lanes 16-31
    const int l16  = lane & 15;

    const int mSub = wave & 1;                // 0..1
    const int nSub = wave >> 1;               // 0..3

    const int mRow = mSub * 16 + l16;         // A row within LDS tile
    const int nCol = nBase + nSub * 16 + l16; // global output column

    // Bias folded into accumulator: every C/D element of this lane has col nCol.
    const float bias = bh[nCol];
    v8f acc;
    #pragma unroll
    for (int i = 0; i < 8; ++i) acc[i] = bias;

    #pragma unroll
    for (int k = 0; k < DIN; k += 4) {
        // A fragment (16x4 fp32): lane(m,h) -> {K=k+2h, K=k+2h+1}
        v2f a = *(const v2f*)&As[mRow * A_STRIDE + k + 2 * half];
        // B fragment (4x16 fp32): lane(n,h) -> {K=k+2h, K=k+2h+1} at column nCol
        const float* wcol = wx + (size_t)(k + 2 * half) * DLAT + nCol;
        v2f b;
        b.x = wcol[0];
        b.y = wcol[DLAT];
        acc = __builtin_amdgcn_wmma_f32_16x16x4_f32(
            /*neg_a=*/false, a, /*neg_b=*/false, b,
            /*c_mod=*/(short)0, acc, /*reuse_a=*/false, /*reuse_b=*/false);
    }

    // Store C: VGPR v, half h -> row = v + 8h, col = l16.
    float* Urow = U + (size_t)(mBase + mSub * 16) * DLAT + nBase + nSub * 16;
    #pragma unroll
    for (int v = 0; v < 8; ++v) {
        const int row = v + 8 * half;
        Urow[(size_t)row * DLAT + l16] = acc[v];
    }
}

// ---------------------------------------------------------------------------
// Kernel 2: per-(chunk, channel) local scan with zero init -> carry.
// idx layout: channel fastest (coalesced across the 16384-float U rows).
// ---------------------------------------------------------------------------
__global__ __launch_bounds__(256)
void rnn_scan_carry(const float* __restrict__ U,
                    const float* __restrict__ wh,
                    float* __restrict__ carry) {
    const int idx     = blockIdx.x * blockDim.x + threadIdx.x; // < NCHUNK*CH
    const int channel = idx & (CH - 1);
    const int chunk   = idx >> 14;            // CH = 2^14
    const float w     = wh[channel & (DLAT - 1)];

    const float* Up = U + (size_t)chunk * CHUNK * CH + channel;
    float h = 0.0f;
    #pragma unroll 8
    for (int t = 0; t < CHUNK; ++t) {
        h = fmaf(h, w, Up[(size_t)t * CH]);
    }
    carry[idx] = h;                            // [chunk][channel]
}

// ---------------------------------------------------------------------------
// Kernel 3: tiny 32-step prefix over chunk carries: h_start per chunk.
// h_start[c] = wh^CHUNK * h_start[c-1] + carry[c-1]
// ---------------------------------------------------------------------------
__global__ __launch_bounds__(256)
void rnn_scan_prefix(const float* __restrict__ carry,
                     const float* __restrict__ wh,
                     const float* __restrict__ h0,
                     float* __restrict__ hstart) {
    const int channel = blockIdx.x * blockDim.x + threadIdx.x; // < CH
    const float w = wh[channel & (DLAT - 1)];

    float wC = w;                              // wh^128 via 7 squarings
    #pragma unroll
    for (int i = 0; i < 7; ++i) wC *= wC;

    float hs = h0[channel];                    // initial hidden state (zeros)
    hstart[channel] = hs;
    #pragma unroll
    for (int c = 1; c < NCHUNK; ++c) {
        hs = fmaf(hs, wC, carry[(size_t)(c - 1) * CH + channel]);
        hstart[(size_t)c * CH + channel] = hs;
    }
}

// ---------------------------------------------------------------------------
// Kernel 4: re-scan each chunk with its true start; tanh, in-place on d_out.
// ---------------------------------------------------------------------------
__global__ __launch_bounds__(256)
void rnn_scan_final(const float* __restrict__ wh,
                    const float* __restrict__ hstart,
                    float* __restrict__ out) {
    const int idx     = blockIdx.x * blockDim.x + threadIdx.x; // < NCHUNK*CH
    const int channel = idx & (CH - 1);
    const int chunk   = idx >> 14;
    const float w     = wh[channel & (DLAT - 1)];

    float* Up = out + (size_t)chunk * CHUNK * CH + channel;
    float h = hstart[idx];
    #pragma unroll 8
    for (int t = 0; t < CHUNK; ++t) {
        const float u = Up[(size_t)t * CH];
        h = fmaf(h, w, u);
        Up[(size_t)t * CH] = tanhf(h);
    }
}

// ---------------------------------------------------------------------------
extern "C" void kernel_launch(void* const* d_in, const int* in_sizes, int n_in,
                              void* d_out, int out_size, void* d_ws, size_t ws_size,
                              hipStream_t stream) {
    const float* X  = (const float*)d_in[0];  // [L, B, 128]
    const float* h0 = (const float*)d_in[1];  // [B, 512]
    const float* wx = (const float*)d_in[2];  // [128, 512]
    const float* wh = (const float*)d_in[3];  // [512]
    const float* bh = (const float*)d_in[4];  // [512]
    float* out = (float*)d_out;               // [L, B, 512] -- doubles as U

    float* carry  = (float*)d_ws;             // [NCHUNK, CH]  = 2 MB
    float* hstart = carry + (size_t)NCHUNK * CH; // [NCHUNK, CH] = 2 MB

    // 1) U = X @ wx + bh  (into d_out)
    dim3 g1(MTOT / 32, DLAT / 64);            // 4096 x 8 blocks
    rnn_gemm_wmma<<<g1, 256, 0, stream>>>(X, wx, bh, out);

    // 2) per-chunk local scans -> carries
    const int n2 = NCHUNK * CH;               // 524288 threads
    rnn_scan_carry<<<n2 / 256, 256, 0, stream>>>(out, wh, carry);

    // 3) prefix over chunks -> per-chunk start states
    rnn_scan_prefix<<<CH / 256, 256, 0, stream>>>(carry, wh, h0, hstart);

    // 4) final scan + tanh, in-place
    rnn_scan_final<<<n2 / 256, 256, 0, stream>>>(wh, hstart, out);
}